// DecodeBoxScript_89240830476328
// MI455X (gfx1250) — compile-verified
//
#include <hip/hip_runtime.h>
#include <hip/hip_bf16.h>
#include <math.h>

// ---------------------------------------------------------------------------
// Problem constants (from reference)
// ---------------------------------------------------------------------------
#define BB     32
#define AA     3
#define HH     52
#define WW     52
#define HWW    (HH*WW)          // 2704
#define ATT    85               // 4 box + 1 conf + 80 cls
#define ROWS   (AA*HWW)         // 8112 rows per batch
#define OUT_ELEMS ((size_t)BB * ROWS * ATT)   // 22,064,640
#define MAXDET 100
#define NEGV   (-1.0e9f)

#define TILE_SP 128             // spatial positions per block (4 wave32 waves)
#define NTILES  22              // ceil(2704/128)

// ---------------------------------------------------------------------------
// CDNA5 async LDS->global DMA path (guarded; falls back to plain stores)
// Builtin signature (from hipcc diagnostic): (v4i __device__*, v4i __shared__*, Ii, Ii)
// ---------------------------------------------------------------------------
#if defined(__AMDGCN__) && __has_builtin(__builtin_amdgcn_global_store_async_from_lds_b128)
#define HAVE_ASYNC_STORE 1
typedef int v4i __attribute__((ext_vector_type(4)));
typedef __attribute__((address_space(1))) v4i* g4p;   // global int4*
typedef __attribute__((address_space(3))) v4i* l4p;   // LDS int4*
#endif

__device__ __forceinline__ float sigm(float x) {
    return 1.0f / (1.0f + expf(-x));
}

// ---------------------------------------------------------------------------
// Kernel 1: decode.  One block = one (batch, anchor, spatial-tile).
// Loads are plane-major (fully coalesced: lane == spatial), transforms applied
// in registers, scattered into an LDS tile laid out exactly like the output
// (row-major [spatial][attr]), then the tile is DMA'd LDS->global with
// GLOBAL_STORE_ASYNC_FROM_LDS_B128 (16B lanes, region is 16B aligned).
// LDS write stride 85 floats: 85 mod 64 = 21, coprime with 64 -> conflict-free.
// ---------------------------------------------------------------------------
__global__ __launch_bounds__(TILE_SP) void decode_kernel(
    const float* __restrict__ in,       // (32,255,52,52)
    const float* __restrict__ anchors,  // (9,2)
    float* __restrict__ out)            // (32,8112,85)
{
    __shared__ __align__(16) float tile[TILE_SP * ATT];

    const int t    = threadIdx.x;
    const int tix  = blockIdx.x;        // spatial tile
    const int a    = blockIdx.y;        // anchor (mask0 -> anchors[6+a])
    const int b    = blockIdx.z;        // batch
    const int s0   = tix * TILE_SP;
    const int s    = s0 + t;

    // anc = anchors[6+a] / stride(=8); boxes later divided by 52.
    const float invW = 1.0f / 52.0f;
    const float ancw = anchors[(6 + a) * 2 + 0] * 0.125f * invW;
    const float anch = anchors[(6 + a) * 2 + 1] * 0.125f * invW;

    const float* base = in + ((size_t)b * 255 + (size_t)a * ATT) * HWW;

    if (s < HWW) {
        const float gx = (float)(s % WW);
        const float gy = (float)(s / WW);
        float* row = &tile[t * ATT];

        row[0] = (sigm(base[0 * HWW + s]) + gx) * invW;   // bx / W
        row[1] = (sigm(base[1 * HWW + s]) + gy) * invW;   // by / H
        row[2] = expf(base[2 * HWW + s]) * ancw;          // bw / W
        row[3] = expf(base[3 * HWW + s]) * anch;          // bh / H
        row[4] = sigm(base[4 * HWW + s]);                 // conf
#pragma unroll 4
        for (int c = 5; c < ATT; ++c)
            row[c] = sigm(base[c * HWW + s]);             // 80 classes
    }
    __syncthreads();

    // Contiguous flat copy: LDS tile layout == output layout for this tile.
    const int nsp = (HWW - s0) < TILE_SP ? (HWW - s0) : TILE_SP;   // 128 or 16
    const int nf  = nsp * ATT;                                      // floats
    float* dst = out + ((size_t)b * ROWS + (size_t)a * HWW + s0) * ATT;

#if HAVE_ASYNC_STORE
    // nf*4 bytes is always a multiple of 16 (128*85*4, 16*85*4); region start
    // is 16B aligned (s0 multiple of 128, 128*85*4 % 16 == 0).
    const int nvec = nf >> 2;                       // 16-byte units
    for (int i = t; i < nvec; i += TILE_SP) {
        __builtin_amdgcn_global_store_async_from_lds_b128(
            (g4p)(dst + i * 4), (l4p)(tile + i * 4), 0, 0);
    }
#if __has_builtin(__builtin_amdgcn_s_wait_asynccnt)
    __builtin_amdgcn_s_wait_asynccnt(0);
#endif
#else
    for (int i = t; i < nf; i += TILE_SP) dst[i] = tile[i];
#endif
}

// ---------------------------------------------------------------------------
// Kernel 2: per-row NMS prep on batch 0.
// ws layout (floats): obox[8112*4] | corners[8112*4] | scores[8112] |
//                     conf[8112]   | clsconf[8112]   | clspred[8112]
// ---------------------------------------------------------------------------
__global__ void prep_kernel(const float* __restrict__ out0, float* __restrict__ ws)
{
    const int r = blockIdx.x * blockDim.x + threadIdx.x;
    if (r >= ROWS) return;

    const float* p = out0 + (size_t)r * ATT;
    const float cx = p[0], cy = p[1], w = p[2], h = p[3], cf = p[4];

    float best = p[5];
    int   bc   = 0;
    for (int c = 1; c < 80; ++c) {          // first-max tie-break == jnp.argmax
        float v = p[5 + c];
        if (v > best) { best = v; bc = c; }
    }
    const float score = cf * best;

    const float x1 = cx - 0.5f * w, y1 = cy - 0.5f * h;
    const float x2 = cx + 0.5f * w, y2 = cy + 0.5f * h;
    const float off = (float)bc * 4096.0f;

    float* obox    = ws;
    float* corners = ws + 4 * ROWS;
    float* scr     = ws + 8 * ROWS;
    float* conf    = ws + 9 * ROWS;
    float* ccf     = ws + 10 * ROWS;
    float* cpd     = ws + 11 * ROWS;

    obox[r * 4 + 0] = x1 + off;  obox[r * 4 + 1] = y1 + off;
    obox[r * 4 + 2] = x2 + off;  obox[r * 4 + 3] = y2 + off;
    corners[r * 4 + 0] = x1;  corners[r * 4 + 1] = y1;
    corners[r * 4 + 2] = x2;  corners[r * 4 + 3] = y2;
    scr[r]  = (score >= 0.5f) ? score : NEGV;
    conf[r] = cf;  ccf[r] = best;  cpd[r] = (float)bc;
}

// ---------------------------------------------------------------------------
// Kernel 3: sequential NMS scan (100 steps), single block of 1024 threads
// (32 wave32 waves).  Scores live in LDS; argmax via wave32 shfl_xor reduction
// with lowest-index tie-break (matches jnp.argmax on all-NEG plateau).
// ---------------------------------------------------------------------------
__global__ __launch_bounds__(1024) void nms_kernel(
    const float* __restrict__ ws, float* __restrict__ det)
{
    __shared__ float sS[ROWS];              // 32448 B
    __shared__ float wv[32];
    __shared__ int   wi[32];
    __shared__ float sbox[4];
    __shared__ int   sIdx, sFlag;
    __shared__ int   idxs[MAXDET];
    __shared__ float flg[MAXDET];

    const float* obox    = ws;
    const float* corners = ws + 4 * ROWS;
    const float* scr0    = ws + 8 * ROWS;
    const float* conf    = ws + 9 * ROWS;
    const float* ccf     = ws + 10 * ROWS;
    const float* cpd     = ws + 11 * ROWS;

    const int tid  = threadIdx.x;
    const int lane = tid & 31;
    const int wave = tid >> 5;

    for (int j = tid; j < ROWS; j += 1024) sS[j] = scr0[j];
    __syncthreads();

    for (int it = 0; it < MAXDET; ++it) {
        // ---- block argmax (value desc, index asc tie-break) ----
        float bv = -3.0e9f;
        int   bi = 0x7FFFFFFF;
        for (int j = tid; j < ROWS; j += 1024) {
            float v = sS[j];
            if (v > bv) { bv = v; bi = j; }
        }
        for (int off = 16; off > 0; off >>= 1) {
            float ov = __shfl_xor(bv, off, 32);
            int   oi = __shfl_xor(bi, off, 32);
            if (ov > bv || (ov == bv && oi < bi)) { bv = ov; bi = oi; }
        }
        if (lane == 0) { wv[wave] = bv; wi[wave] = bi; }
        __syncthreads();
        if (tid < 32) {
            bv = wv[tid]; bi = wi[tid];
            for (int off = 16; off > 0; off >>= 1) {
                float ov = __shfl_xor(bv, off, 32);
                int   oi = __shfl_xor(bi, off, 32);
                if (ov > bv || (ov == bv && oi < bi)) { bv = ov; bi = oi; }
            }
            if (tid == 0) {
                sIdx  = bi;
                sFlag = (sS[bi] > -5.0e8f) ? 1 : 0;
                sbox[0] = obox[bi * 4 + 0];  sbox[1] = obox[bi * 4 + 1];
                sbox[2] = obox[bi * 4 + 2];  sbox[3] = obox[bi * 4 + 3];
                idxs[it] = bi;
                flg[it]  = sFlag ? 1.0f : 0.0f;
            }
        }
        __syncthreads();

        // ---- IoU suppression ----
        const float bx1 = sbox[0], by1 = sbox[1], bx2 = sbox[2], by2 = sbox[3];
        const float a1  = (bx2 - bx1) * (by2 - by1);
        const int   ii  = sIdx;
        for (int j = tid; j < ROWS; j += 1024) {
            const float* o = &obox[j * 4];
            float lx = fmaxf(bx1, o[0]), ly = fmaxf(by1, o[1]);
            float rx = fminf(bx2, o[2]), ry = fminf(by2, o[3]);
            float iw = fmaxf(rx - lx, 0.0f), ih = fmaxf(ry - ly, 0.0f);
            float inter = iw * ih;
            float a2 = (o[2] - o[0]) * (o[3] - o[1]);
            float iou = inter / (a1 + a2 - inter + 1e-9f);
            if (iou > 0.4f || j == ii) sS[j] = NEGV;
        }
        __syncthreads();
    }

    // ---- emit det (100 x 7): [y1,x1,y2,x2]*416, conf, clsconf, clspred, *flag
    if (tid < MAXDET) {
        const int   i = idxs[tid];
        const float f = flg[tid];
        const float x1 = corners[i * 4 + 0], y1 = corners[i * 4 + 1];
        const float x2 = corners[i * 4 + 2], y2 = corners[i * 4 + 3];
        float* d = det + tid * 7;
        d[0] = y1 * 416.0f * f;
        d[1] = x1 * 416.0f * f;
        d[2] = y2 * 416.0f * f;
        d[3] = x2 * 416.0f * f;
        d[4] = conf[i] * f;
        d[5] = ccf[i] * f;
        d[6] = cpd[i] * f;
    }
}

// ---------------------------------------------------------------------------
// Launch
// ---------------------------------------------------------------------------
extern "C" void kernel_launch(void* const* d_in, const int* in_sizes, int n_in,
                              void* d_out, int out_size, void* d_ws, size_t ws_size,
                              hipStream_t stream) {
    const float* in      = (const float*)d_in[0];
    const float* anchors = (const float*)d_in[1];
    float* out = (float*)d_out;
    float* det = out + OUT_ELEMS;          // tuple output: out (flat) then det
    float* ws  = (float*)d_ws;             // needs 12*8112 floats (~390 KB)

    dim3 g(NTILES, AA, BB);
    decode_kernel<<<g, TILE_SP, 0, stream>>>(in, anchors, out);
    prep_kernel<<<(ROWS + 255) / 256, 256, 0, stream>>>(out, ws);   // batch 0
    nms_kernel<<<1, 1024, 0, stream>>>(ws, det);
}